// HeterConv_4037269258336
// MI455X (gfx1250) — compile-verified
//
#include <hip/hip_runtime.h>

#define B_  4
#define M_  8192
#define NS_ 8192
#define K_  8
#define F_  288
#define NN_ (M_ + NS_)

typedef __attribute__((ext_vector_type(2))) float v2f;
typedef __attribute__((ext_vector_type(8))) float v8f;

// ---------- degree histogram -> dnorm = deg_out^-1/2 ----------
__global__ void k_zero_deg(int* __restrict__ deg) {
    int i = blockIdx.x * blockDim.x + threadIdx.x;
    if (i < B_ * NN_) deg[i] = 0;
}

__global__ void k_hist(const int* __restrict__ idx_k1,
                       const int* __restrict__ idx_k2,
                       int* __restrict__ deg) {
    int i = blockIdx.x * blockDim.x + threadIdx.x;
    if (i >= B_ * M_ * K_) return;
    int b = i / (M_ * K_);
    atomicAdd(&deg[b * NN_ + idx_k2[i]], 1);        // content sources
    atomicAdd(&deg[b * NN_ + M_ + idx_k1[i]], 1);   // style sources
}

__global__ void k_rsqrt(float* __restrict__ dnorm) {
    int i = blockIdx.x * blockDim.x + threadIdx.x;
    if (i >= B_ * NN_) return;
    int c = ((const int*)dnorm)[i];
    dnorm[i] = rsqrtf((float)(c < 1 ? 1 : c));
}

// ---------- layer-1 aggregation: agg[r,t] = 0.25 * sum_j dn(u_j)*feat[u_j,t] ----------
__global__ __launch_bounds__(F_) void k_agg1(const float* __restrict__ feat_c,
                                             const float* __restrict__ feat_s,
                                             const int* __restrict__ idx_k1,
                                             const int* __restrict__ idx_k2,
                                             const float* __restrict__ dnorm,
                                             float* __restrict__ agg) {
    __shared__ int   su[2 * K_];
    __shared__ float sw[2 * K_];
    int r = blockIdx.x;          // b*M + v
    int t = threadIdx.x;         // feature index 0..287
    int b = r / M_;
    const float* dn = dnorm + (size_t)b * NN_;
    if (t < K_) {
        int u = idx_k2[(size_t)r * K_ + t];
        su[t] = u;           sw[t] = dn[u];
    } else if (t < 2 * K_) {
        int j = t - K_;
        int u = idx_k1[(size_t)r * K_ + j];
        su[t] = u;           sw[t] = dn[M_ + u];
    }
    __syncthreads();
    float acc = 0.f;
#pragma unroll
    for (int j = 0; j < K_; ++j)
        acc += sw[j] * feat_c[((size_t)b * M_ + su[j]) * F_ + t];
#pragma unroll
    for (int j = 0; j < K_; ++j)
        acc += sw[K_ + j] * feat_s[((size_t)b * NS_ + su[K_ + j]) * F_ + t];
    agg[(size_t)r * F_ + t] = acc * 0.25f;   // deg_in^-1/2 = 16^-1/2 ... squared? no: 1/sqrt(16)=0.25
}

// ---------- layer-2 aggregation: content gathers from h1, style term is analytic ----------
__global__ __launch_bounds__(F_) void k_agg2(const float* __restrict__ h1,
                                             const float* __restrict__ bias1,
                                             const int* __restrict__ idx_k1,
                                             const int* __restrict__ idx_k2,
                                             const float* __restrict__ dnorm,
                                             float* __restrict__ agg) {
    __shared__ int   su[K_];
    __shared__ float sw[K_];
    __shared__ float ssum_sh;
    int r = blockIdx.x;
    int t = threadIdx.x;
    int b = r / M_;
    const float* dn = dnorm + (size_t)b * NN_;
    if (t < K_) {
        int u = idx_k2[(size_t)r * K_ + t];
        su[t] = u;  sw[t] = dn[u];
    } else if (t == K_) {
        float s = 0.f;
#pragma unroll
        for (int j = 0; j < K_; ++j)
            s += dn[M_ + idx_k1[(size_t)r * K_ + j]];
        ssum_sh = s;
    }
    __syncthreads();
    float acc = 0.f;
#pragma unroll
    for (int j = 0; j < K_; ++j)
        acc += sw[j] * h1[((size_t)b * M_ + su[j]) * F_ + t];
    // all style nodes carry the constant layer-1 row relu(b1)
    acc += fmaxf(bias1[t], 0.f) * ssum_sh;
    agg[(size_t)r * F_ + t] = acc * 0.25f;
}

// ---------- WMMA fp32 GEMM: out[R,F] = act(A[R,F] @ W[F,F] + bias) ----------
template <bool RELU>
__global__ __launch_bounds__(256) void k_gemm(const float* __restrict__ A,
                                              const float* __restrict__ W,
                                              const float* __restrict__ bias,
                                              float* __restrict__ out) {
    __shared__ float As[16 * F_];                 // 18 KB strip of A
    const int tid     = threadIdx.x;
    const int rowBase = blockIdx.x * 16;

    // cooperative load of 16 x 288 A-strip into LDS
    for (int i = tid; i < 16 * F_; i += 256) {
        int rr = i / F_, cc = i - rr * F_;
        As[i] = A[((size_t)rowBase + rr) * F_ + cc];
    }
    __syncthreads();

    const int wave = tid >> 5;
    const int lane = tid & 31;
    const int c0   = blockIdx.y * 128 + wave * 16;   // 16-wide column tile per wave
    if (c0 >= F_) return;                            // wave-uniform exit (EXEC stays all-1s)

    const int half = lane >> 4;                      // K split across lane halves
    const int lm   = lane & 15;
    const int col  = c0 + lm;

    v8f acc = {};
    for (int k0 = 0; k0 < F_; k0 += 4) {
        // A 16x4 fragment (ISA 7.12.2): lanes 0-15 -> K=k0,k0+1 ; lanes 16-31 -> K=k0+2,k0+3
        v2f a, bf;
        const float* ap = &As[lm * F_ + k0 + 2 * half];
        a.x = ap[0];
        a.y = ap[1];
        // B 4x16 fragment: row-major W, mirrored K layout
        const float* wp = W + (size_t)(k0 + 2 * half) * F_ + col;
        __builtin_prefetch(wp + 4 * F_, 0, 1);       // global_prefetch_b8, next K-step
        bf.x = wp[0];
        bf.y = wp[F_];
        acc = __builtin_amdgcn_wmma_f32_16x16x4_f32(
                  false, a, false, bf, (short)0, acc, false, false);
    }

    const float bv = bias[col];
#pragma unroll
    for (int r = 0; r < 8; ++r) {                    // C/D: VGPR r -> row r (+8 for lane half)
        float v = acc[r] + bv;
        if (RELU) v = fmaxf(v, 0.f);
        out[((size_t)rowBase + r + 8 * half) * F_ + col] = v;
    }
}

extern "C" void kernel_launch(void* const* d_in, const int* in_sizes, int n_in,
                              void* d_out, int out_size, void* d_ws, size_t ws_size,
                              hipStream_t stream) {
    (void)in_sizes; (void)n_in; (void)out_size; (void)ws_size;

    const float* feat_c = (const float*)d_in[0];
    const float* feat_s = (const float*)d_in[1];
    const int*   idx_k1 = (const int*)d_in[2];
    const int*   idx_k2 = (const int*)d_in[3];
    const float* W1     = (const float*)d_in[4];
    const float* b1     = (const float*)d_in[5];
    const float* W2     = (const float*)d_in[6];
    const float* b2     = (const float*)d_in[7];
    float*       out    = (float*)d_out;

    // workspace layout: dnorm | bufA (agg) | bufB (h1)
    float* dnorm = (float*)d_ws;
    float* bufA  = dnorm + (size_t)B_ * NN_;
    float* bufB  = bufA + (size_t)B_ * M_ * F_;

    const int ndeg = B_ * NN_;
    const int nedg = B_ * M_ * K_;
    k_zero_deg<<<(ndeg + 255) / 256, 256, 0, stream>>>((int*)dnorm);
    k_hist   <<<(nedg + 255) / 256, 256, 0, stream>>>(idx_k1, idx_k2, (int*)dnorm);
    k_rsqrt  <<<(ndeg + 255) / 256, 256, 0, stream>>>(dnorm);

    dim3 gGemm(B_ * M_ / 16, (F_ + 127) / 128);      // 2048 x 3

    k_agg1<<<B_ * M_, F_, 0, stream>>>(feat_c, feat_s, idx_k1, idx_k2, dnorm, bufA);
    k_gemm<true ><<<gGemm, 256, 0, stream>>>(bufA, W1, b1, bufB);      // h1 = relu(agg@W1+b1)
    k_agg2<<<B_ * M_, F_, 0, stream>>>(bufB, b1, idx_k1, idx_k2, dnorm, bufA);
    k_gemm<false><<<gGemm, 256, 0, stream>>>(bufA, W2, b2, out);       // out = agg2@W2+b2
}